// ExceptionGateBank_87814901334225
// MI455X (gfx1250) — compile-verified
//
#include <hip/hip_runtime.h>

typedef __attribute__((ext_vector_type(2))) float v2f;
typedef __attribute__((ext_vector_type(8))) float v8f;

#define EXC_DIM 64
#define DST_DIM 128

// ---------------------------------------------------------------------------
// Kernel 0: zero the accumulator region of the workspace (sums, counts, gsum)
// ---------------------------------------------------------------------------
__global__ __launch_bounds__(256)
void zero_f4_kernel(float4* __restrict__ p, long n4) {
    long i = (long)blockIdx.x * blockDim.x + threadIdx.x;
    long stride = (long)gridDim.x * blockDim.x;
    float4 z = {0.f, 0.f, 0.f, 0.f};
    for (; i < n4; i += stride) p[i] = z;
}

// ---------------------------------------------------------------------------
// Kernel 1: row softmax over 64 classes + block-reduced global sum.
// 16 lanes per row, float4 per lane, 2 rows per wave32, 16 rows per block.
// ---------------------------------------------------------------------------
__global__ __launch_bounds__(256)
void softmax_kernel(const float* __restrict__ logits, float* __restrict__ p_exc,
                    float* __restrict__ gsum, int n_exc) {
    __shared__ float lsum[EXC_DIM];
    const int tid = threadIdx.x;
    if (tid < EXC_DIM) lsum[tid] = 0.0f;
    __syncthreads();

    const int rowInBlk = tid >> 4;           // 0..15
    const int l        = tid & 15;           // 16-lane group position
    const int row      = blockIdx.x * 16 + rowInBlk;
    if (row < n_exc) {
        float4 v = ((const float4*)(logits + (size_t)row * EXC_DIM))[l];
        float mx = fmaxf(fmaxf(v.x, v.y), fmaxf(v.z, v.w));
        #pragma unroll
        for (int m = 1; m < 16; m <<= 1) mx = fmaxf(mx, __shfl_xor(mx, m, 32));
        float4 e;
        e.x = __expf(v.x - mx); e.y = __expf(v.y - mx);
        e.z = __expf(v.z - mx); e.w = __expf(v.w - mx);
        float s = e.x + e.y + e.z + e.w;
        #pragma unroll
        for (int m = 1; m < 16; m <<= 1) s += __shfl_xor(s, m, 32);
        float r = 1.0f / s;
        float4 p; p.x = e.x * r; p.y = e.y * r; p.z = e.z * r; p.w = e.w * r;
        ((float4*)(p_exc + (size_t)row * EXC_DIM))[l] = p;
        // per-dim partials for the global mean (LDS f32 atomics)
        atomicAdd(&lsum[l * 4 + 0], p.x);
        atomicAdd(&lsum[l * 4 + 1], p.y);
        atomicAdd(&lsum[l * 4 + 2], p.z);
        atomicAdd(&lsum[l * 4 + 3], p.w);
    }
    __syncthreads();
    if (tid < EXC_DIM) atomicAdd(&gsum[tid], lsum[tid]);
}

// ---------------------------------------------------------------------------
// Kernel 2: edge scatter-add. 16 lanes per edge; p_exc & sums are L2-resident.
// ---------------------------------------------------------------------------
__global__ __launch_bounds__(256)
void scatter_kernel(const float* __restrict__ p_exc, const int* __restrict__ src_idx,
                    const int* __restrict__ dst_idx, float* __restrict__ sums,
                    float* __restrict__ counts, int n_edges) {
    long tid = (long)blockIdx.x * blockDim.x + threadIdx.x;
    int e = (int)(tid >> 4);
    int l = (int)(tid & 15);
    if (e >= n_edges) return;
    int s = src_idx[e];
    int d = dst_idx[e];
    float4 p = ((const float4*)(p_exc + (size_t)s * EXC_DIM))[l];
    float* srow = sums + (size_t)d * EXC_DIM + l * 4;
    atomicAdd(srow + 0, p.x);
    atomicAdd(srow + 1, p.y);
    atomicAdd(srow + 2, p.z);
    atomicAdd(srow + 3, p.w);
    if (l == 0) atomicAdd(&counts[d], 1.0f);
}

// ---------------------------------------------------------------------------
// Kernel 3: finalize with WMMA. One wave32 per 16 dst rows.
// P[16x64] @ u[64] via 16 steps of V_WMMA_F32_16X16X4_F32 (B = u replicated
// across all N columns, so every column of D holds the 16 dot products).
// A f32 16x4 layout: lane m (0..15) fragment = (A[m][2g], A[m][2g+1]), g=lane>>4.
// ---------------------------------------------------------------------------
__global__ __launch_bounds__(32)
void finalize_kernel(const float* __restrict__ sums, const float* __restrict__ counts,
                     const float* __restrict__ gsum, const float* __restrict__ u,
                     const float* __restrict__ m_rule, float* __restrict__ out,
                     int n_exc) {
    __shared__ float gdot[16];
    const int lane = threadIdx.x;        // 0..31
    const int m    = lane & 15;
    const int g    = lane >> 4;          // half-wave id (K pair selector)
    const int dstBase = blockIdx.x * 16;
    const int row  = dstBase + m;

    const float cnt  = counts[row];
    const float inv  = 1.0f / fmaxf(cnt, 1.0f);
    const bool  has  = cnt > 0.0f;
    const float invN = 1.0f / (float)n_exc;

    const float* srow = sums + (size_t)row * EXC_DIM;

    v8f acc = {};
    #pragma unroll
    for (int c = 0; c < 16; ++c) {
        const int off = c * 4 + 2 * g;
        float s0 = srow[off + 0];
        float s1 = srow[off + 1];
        float g0 = gsum[off + 0] * invN;
        float g1 = gsum[off + 1] * invN;
        v2f a, b;
        a.x = has ? s0 * inv : g0;       // select, no EXEC divergence
        a.y = has ? s1 * inv : g1;
        b.x = u[off + 0];                // replicated over N columns
        b.y = u[off + 1];
        acc = __builtin_amdgcn_wmma_f32_16x16x4_f32(
            /*neg_a=*/false, a, /*neg_b=*/false, b,
            /*c_mod=*/(short)0, acc, /*reuse_a=*/false, /*reuse_b=*/false);
    }

    // D layout: VGPR r, lanes 0-15 -> M=r; lanes 16-31 -> M=r+8 (any column).
    if (lane == 0) {
        #pragma unroll
        for (int r = 0; r < 8; ++r) gdot[r] = acc[r];
    }
    if (lane == 16) {
        #pragma unroll
        for (int r = 0; r < 8; ++r) gdot[8 + r] = acc[r];
    }
    __syncthreads();

    // 16 rows x 128 dims: 32 float4 per row, one per lane -> coalesced.
    #pragma unroll
    for (int rl = 0; rl < 16; ++rl) {
        float dot  = gdot[rl];
        float sg   = 1.0f / (1.0f + __expf(-dot));
        float gate = 1.0f - sg;
        const float4* mr = (const float4*)(m_rule + (size_t)(dstBase + rl) * DST_DIM);
        float4*       op = (float4*)(out + (size_t)(dstBase + rl) * DST_DIM);
        float4 v = mr[lane];
        v.x *= gate; v.y *= gate; v.z *= gate; v.w *= gate;
        op[lane] = v;
    }
}

// ---------------------------------------------------------------------------
// Launch
// ---------------------------------------------------------------------------
extern "C" void kernel_launch(void* const* d_in, const int* in_sizes, int n_in,
                              void* d_out, int out_size, void* d_ws, size_t ws_size,
                              hipStream_t stream) {
    const float* logits  = (const float*)d_in[0];   // [n_exc, 64]
    const float* m_rule  = (const float*)d_in[1];   // [n_dst, 128]
    const float* u       = (const float*)d_in[2];   // [64]
    const int*   src_idx = (const int*)d_in[3];     // [n_edges]
    const int*   dst_idx = (const int*)d_in[4];     // [n_edges]
    float*       out     = (float*)d_out;

    const int n_exc   = in_sizes[0] / EXC_DIM;      // 200000
    const int n_dst   = in_sizes[1] / DST_DIM;      // 100000
    const int n_edges = in_sizes[3];                // 2000000

    // workspace layout (floats): p_exc | sums | counts | gsum
    float* ws     = (float*)d_ws;
    float* p_exc  = ws;                                   // n_exc * 64
    float* sums   = p_exc + (size_t)n_exc * EXC_DIM;      // n_dst * 64
    float* counts = sums + (size_t)n_dst * EXC_DIM;       // n_dst
    float* gsum   = counts + n_dst;                       // 64

    // 0) zero sums + counts + gsum (contiguous, multiple of 4 floats)
    long nZero  = (long)n_dst * EXC_DIM + n_dst + EXC_DIM;
    long nZero4 = (nZero + 3) / 4;
    int zGrid = (int)((nZero4 + 255) / 256);
    if (zGrid > 4096) zGrid = 4096;
    zero_f4_kernel<<<zGrid, 256, 0, stream>>>((float4*)sums, nZero4);

    // 1) softmax + global-mean partials (16 rows per 256-thread block)
    int smGrid = (n_exc + 15) / 16;
    softmax_kernel<<<smGrid, 256, 0, stream>>>(logits, p_exc, gsum, n_exc);

    // 2) edge scatter (16 lanes per edge)
    long scThreads = (long)n_edges * 16;
    int scGrid = (int)((scThreads + 255) / 256);
    scatter_kernel<<<scGrid, 256, 0, stream>>>(p_exc, src_idx, dst_idx, sums, counts, n_edges);

    // 3) WMMA gate + output (one wave per 16 dst rows; n_dst % 16 == 0)
    int fGrid = n_dst / 16;
    finalize_kernel<<<fGrid, 32, 0, stream>>>(sums, counts, gsum, u, m_rule, out, n_exc);
}